// DigitCaps_52132313039390
// MI455X (gfx1250) — compile-verified
//
#include <hip/hip_runtime.h>
#include <math.h>

// ---------------------------------------------------------------------------
// DigitCaps winner-take-all routing, fused for MI455X (gfx1250, wave32, WMMA)
//
// Shapes (fixed by reference setup_inputs):
//   x : [64, 14,14,32, 8]  -> [B=64, N=6272, DIN=8]
//   W : [6272, 8, 160]          (d*m = 16*10)
//   dc: [10, 16]
//   out: [64, 10] float32
//
// Per n (6272 workgroups, 4 waves, one 16-batch tile each), per capsule mm:
//   U^T  = W_n^T(16x8) @ X_n^T(8x16)        : 2x V_WMMA_F32_16X16X4_F32
//   squash in D-layout (8 FMA + 1 xor16 + v_sqrt/v_tanh/v_rcp)
//   S    = DCpad(16x16) @ U^T(16x16)        : 4x V_WMMA_F32_16X16X4_F32
//   pass1: 3-round WTA + class accumulation;  pass2: sim reduction vs dc_new
// ---------------------------------------------------------------------------

typedef float v2f __attribute__((ext_vector_type(2)));
typedef float v8f __attribute__((ext_vector_type(8)));

namespace cfg {
constexpr int kB    = 64;
constexpr int kN    = 14 * 14 * 32;   // 6272
constexpr int kDin  = 8;
constexpr int kD    = 16;
constexpr int kM    = 10;
constexpr int kC    = 10;
constexpr int kDM   = kD * kM;        // 160
constexpr float kTot   = (float)kB * (float)kN * (float)kM;  // update mean divisor
constexpr float kTotNm = (float)kN * (float)kM;              // sim mean divisor
// workspace float offsets
constexpr int kAccOff = 0;      // [10*16] sum_{b,nm} theta * u per class
constexpr int kCntOff = 160;    // [10]    sum theta (weighted winner counts)
constexpr int kDcnOff = 176;    // [10*16] updated+normalized codebook
constexpr int kSimOff = 336;    // [64*10] sum over nm of u . dc_new
constexpr int kWsTot  = 976;
}

__device__ __forceinline__ float x16(float v) { return __shfl_xor(v, 16, 32); }

__device__ __forceinline__ float fast_sqrtf(float x) {
#if __has_builtin(__builtin_amdgcn_sqrtf)
  return __builtin_amdgcn_sqrtf(x);
#else
  return sqrtf(x);
#endif
}
__device__ __forceinline__ float fast_tanhf(float x) {
#if __has_builtin(__builtin_amdgcn_tanhf)
  return __builtin_amdgcn_tanhf(x);
#else
  return tanhf(x);
#endif
}
__device__ __forceinline__ float fast_rcpf(float x) {
#if __has_builtin(__builtin_amdgcn_rcpf)
  return __builtin_amdgcn_rcpf(x);   // native v_rcp_f32, ~1 ulp
#else
  return 1.0f / x;
#endif
}

__global__ void caps_init(float* __restrict__ ws) {
  for (int i = threadIdx.x; i < cfg::kWsTot; i += blockDim.x) ws[i] = 0.0f;
}

// f32 16x16x4 VGPR layouts (ISA 7.12.2):
//   A: lane L holds rows M=L%16, K = 2*(L/16)+j in VGPR j
//   B: (symmetric) lane L holds cols N=L%16, K = 2*(L/16)+j in VGPR j
//   C/D: lane L, VGPR v -> M = v + 8*(L/16), N = L%16

// U^T tile for capsule mm: A = W_n^T (16x4 chunks), B = X_n^T (4x16 chunks).
__device__ __forceinline__ v8f caps_tile(const float* __restrict__ sW,
                                         v2f b0, v2f b1, int mm,
                                         int col, int hi) {
  const int row = mm * 16 + col;     // row in 160-wide d*m axis
  const int kb  = 2 * hi;
  v2f a0, a1;
  a0.x = sW[(kb + 0) * cfg::kDM + row];
  a0.y = sW[(kb + 1) * cfg::kDM + row];
  a1.x = sW[(kb + 4) * cfg::kDM + row];
  a1.y = sW[(kb + 5) * cfg::kDM + row];
  v8f acc = {};
  acc = __builtin_amdgcn_wmma_f32_16x16x4_f32(false, a0, false, b0,
                                              (short)0, acc, false, false);
  acc = __builtin_amdgcn_wmma_f32_16x16x4_f32(false, a1, false, b1,
                                              (short)0, acc, false, false);
  return acc;
}

// squash in D-layout: per-column |u| via half sum + one xor16; scale in place.
__device__ __forceinline__ void caps_squash(v8f& acc) {
  float ss = 0.0f;
#pragma unroll
  for (int v = 0; v < 8; ++v) ss = fmaf(acc[v], acc[v], ss);
  ss += x16(ss);
  const float nrm = fast_sqrtf(ss);
  const float sc  = fast_tanhf(nrm) * fast_rcpf(fmaxf(nrm, 1e-20f));
#pragma unroll
  for (int v = 0; v < 8; ++v) acc[v] *= sc;
}

// Repack scaled U tile (D-layout) into four K=4 B-operand chunks.
__device__ __forceinline__ void caps_bchunks(const v8f& acc, int hi, v2f B[4]) {
  float t[8];
#pragma unroll
  for (int v = 0; v < 8; ++v) t[v] = x16(acc[v]);
  const bool h = (hi != 0);
  B[0].x = h ? t[2]   : acc[0];  B[0].y = h ? t[3]   : acc[1];   // d 0..3
  B[1].x = h ? t[6]   : acc[4];  B[1].y = h ? t[7]   : acc[5];   // d 4..7
  B[2].x = h ? acc[2] : t[0];    B[2].y = h ? acc[3] : t[1];     // d 8..11
  B[3].x = h ? acc[6] : t[4];    B[3].y = h ? acc[7] : t[5];     // d 12..15
}

// S = DCpad(16x16, M=c K=d, rows>=10 zero) x U(16x16, d x batch): 4 chained WMMAs.
__device__ __forceinline__ v8f caps_sim(const v2f Adc[4], const v2f B[4]) {
  v8f S = {};
#pragma unroll
  for (int q = 0; q < 4; ++q)
    S = __builtin_amdgcn_wmma_f32_16x16x4_f32(false, Adc[q], false, B[q],
                                              (short)0, S, false, false);
  return S;
}

// Preload codebook into A-layout chunks (zero-padded rows c>=10).
__device__ __forceinline__ void caps_load_dcA(const float* __restrict__ sDC,
                                              int col, int hi, v2f Adc[4]) {
#pragma unroll
  for (int q = 0; q < 4; ++q) {
    const int kq = 4 * q + 2 * hi;
    const bool valid = (col < cfg::kC);
    Adc[q].x = valid ? sDC[col * 16 + kq + 0] : 0.0f;
    Adc[q].y = valid ? sDC[col * 16 + kq + 1] : 0.0f;
  }
}

__device__ __forceinline__ void load_btiles(const float* __restrict__ X, int n,
                                            int wave, int col, int hi,
                                            v2f& b0, v2f& b1) {
  const float* xb = X + (size_t)(wave * 16 + col) * (cfg::kN * cfg::kDin)
                      + (size_t)n * cfg::kDin;
  const int kb = 2 * hi;
  b0.x = xb[kb + 0]; b0.y = xb[kb + 1];
  b1.x = xb[kb + 4]; b1.y = xb[kb + 5];
}

__global__ __launch_bounds__(128)
void caps_pass1(const float* __restrict__ X, const float* __restrict__ W,
                const float* __restrict__ DC, float* __restrict__ ws) {
  __shared__ float sW[cfg::kDin * cfg::kDM];   // 8*160 = 5 KB
  __shared__ float sDC[cfg::kC * cfg::kD];
  __shared__ float sAcc[cfg::kC * cfg::kD];
  __shared__ float sCnt[cfg::kC];

  const int n    = blockIdx.x;
  const int tid  = threadIdx.x;
  const int lane = tid & 31;
  const int wave = tid >> 5;       // batch tile 0..3
  const int col  = lane & 15;
  const int hi   = lane >> 4;

  const float* Wn = W + (size_t)n * (cfg::kDin * cfg::kDM);
  for (int i = tid; i < cfg::kDin * cfg::kDM; i += 128) sW[i] = Wn[i];
  for (int i = tid; i < cfg::kC * cfg::kD; i += 128) { sDC[i] = DC[i]; sAcc[i] = 0.0f; }
  if (tid < cfg::kC) sCnt[tid] = 0.0f;
  if (n + 1 < cfg::kN)  // warm caches for next block's W tile
    __builtin_prefetch(W + (size_t)(n + 1) * (cfg::kDin * cfg::kDM) + tid * 10, 0, 1);
  __syncthreads();

  v2f b0, b1;
  load_btiles(X, n, wave, col, hi, b0, b1);
  v2f Adc[4];
  caps_load_dcA(sDC, col, hi, Adc);

  const float thetas[3] = {1.0f, 0.2f, 0.1f};
#pragma unroll 2
  for (int mm = 0; mm < cfg::kM; ++mm) {
    v8f acc = caps_tile(sW, b0, b1, mm, col, hi);
    caps_squash(acc);
    v2f Bc[4];
    caps_bchunks(acc, hi, Bc);
    v8f S = caps_sim(Adc, Bc);

    // gather all 10 class sims per lane (partner half via xor16)
    float t[8];
#pragma unroll
    for (int v = 0; v < 8; ++v) t[v] = x16(S[v]);
    float s[cfg::kC];
#pragma unroll
    for (int c = 0; c < 8; ++c) s[c] = hi ? t[c] : S[c];
    s[8] = hi ? S[0] : t[0];
    s[9] = hi ? S[1] : t[1];

    // 3-round winner-take-all; each lane accumulates its d-half (= its 8 regs)
#pragma unroll
    for (int r = 0; r < 3; ++r) {
      int w = 0; float best = s[0];
#pragma unroll
      for (int c = 1; c < cfg::kC; ++c) if (s[c] > best) { best = s[c]; w = c; }
      const float th = thetas[r];
#pragma unroll
      for (int v = 0; v < 8; ++v)
        atomicAdd(&sAcc[w * 16 + 8 * hi + v], th * acc[v]);
      if (hi == 0) atomicAdd(&sCnt[w], th);
      s[w] = -3.4e38f;
    }
  }
  __syncthreads();
  for (int i = tid; i < cfg::kC * cfg::kD; i += 128)
    atomicAdd(&ws[cfg::kAccOff + i], sAcc[i]);
  if (tid < cfg::kC) atomicAdd(&ws[cfg::kCntOff + tid], sCnt[tid]);
}

__global__ void caps_update_dc(const float* __restrict__ DC, float* __restrict__ ws) {
  const int c = threadIdx.x;
  if (c >= cfg::kC) return;
  const float cnt = ws[cfg::kCntOff + c];
  float v[16];
  float ss = 0.0f;
#pragma unroll
  for (int d = 0; d < 16; ++d) {
    const float dc0 = DC[c * 16 + d];
    const float up  = (ws[cfg::kAccOff + c * 16 + d] - cnt * dc0) * (1.0f / cfg::kTot);
    const float t   = dc0 + up;
    v[d] = t; ss += t * t;
  }
  const float inv = fast_rcpf(fast_sqrtf(ss));
#pragma unroll
  for (int d = 0; d < 16; ++d) ws[cfg::kDcnOff + c * 16 + d] = v[d] * inv;
}

__global__ __launch_bounds__(128)
void caps_pass2(const float* __restrict__ X, const float* __restrict__ W,
                float* __restrict__ ws) {
  __shared__ float sW[cfg::kDin * cfg::kDM];
  __shared__ float sDC[cfg::kC * cfg::kD];
  __shared__ float sSim[cfg::kB * cfg::kC];

  const int n    = blockIdx.x;
  const int tid  = threadIdx.x;
  const int lane = tid & 31;
  const int wave = tid >> 5;
  const int col  = lane & 15;
  const int hi   = lane >> 4;

  const float* Wn = W + (size_t)n * (cfg::kDin * cfg::kDM);
  for (int i = tid; i < cfg::kDin * cfg::kDM; i += 128) sW[i] = Wn[i];
  for (int i = tid; i < cfg::kC * cfg::kD; i += 128) sDC[i] = ws[cfg::kDcnOff + i];
  __syncthreads();

  v2f b0, b1;
  load_btiles(X, n, wave, col, hi, b0, b1);
  v2f Adc[4];
  caps_load_dcA(sDC, col, hi, Adc);

  v8f sacc = {};
#pragma unroll 2
  for (int mm = 0; mm < cfg::kM; ++mm) {
    v8f acc = caps_tile(sW, b0, b1, mm, col, hi);
    caps_squash(acc);
    v2f Bc[4];
    caps_bchunks(acc, hi, Bc);
    v8f S = caps_sim(Adc, Bc);
#pragma unroll
    for (int v = 0; v < 8; ++v) sacc[v] += S[v];
  }
  // lane owns classes c = v + 8*hi for batch b = wave*16+col; c>=10 discarded
  const int b = wave * 16 + col;
  if (hi == 0) {
#pragma unroll
    for (int v = 0; v < 8; ++v) sSim[b * cfg::kC + v] = sacc[v];
  } else {
    sSim[b * cfg::kC + 8] = sacc[0];
    sSim[b * cfg::kC + 9] = sacc[1];
  }
  __syncthreads();
  for (int i = tid; i < cfg::kB * cfg::kC; i += 128)
    atomicAdd(&ws[cfg::kSimOff + i], sSim[i]);
}

__global__ void caps_softmax(const float* __restrict__ ws, float* __restrict__ out) {
  const int b = threadIdx.x;
  if (b >= cfg::kB) return;
  float v[cfg::kC];
  float mx = -3.4e38f;
#pragma unroll
  for (int c = 0; c < cfg::kC; ++c) {
    v[c] = ws[cfg::kSimOff + b * cfg::kC + c] * (1.0f / cfg::kTotNm);
    mx = fmaxf(mx, v[c]);
  }
  float sum = 0.0f;
#pragma unroll
  for (int c = 0; c < cfg::kC; ++c) { v[c] = __expf(v[c] - mx); sum += v[c]; }
  const float inv = fast_rcpf(sum);
#pragma unroll
  for (int c = 0; c < cfg::kC; ++c) out[b * cfg::kC + c] = v[c] * inv;
}

extern "C" void kernel_launch(void* const* d_in, const int* in_sizes, int n_in,
                              void* d_out, int out_size, void* d_ws, size_t ws_size,
                              hipStream_t stream) {
  const float* X  = (const float*)d_in[0];   // [64,14,14,32,8] f32
  const float* W  = (const float*)d_in[1];   // [6272,8,160]    f32
  const float* DC = (const float*)d_in[2];   // [10,16]         f32
  float* out = (float*)d_out;                // [64,10]         f32
  float* ws  = (float*)d_ws;
  (void)in_sizes; (void)n_in; (void)out_size; (void)ws_size;

  caps_init<<<1, 256, 0, stream>>>(ws);
  caps_pass1<<<cfg::kN, 128, 0, stream>>>(X, W, DC, ws);
  caps_update_dc<<<1, 32, 0, stream>>>(DC, ws);
  caps_pass2<<<cfg::kN, 128, 0, stream>>>(X, W, ws);
  caps_softmax<<<1, 64, 0, stream>>>(ws, out);
}